// NonlinearIB_25744033972501
// MI455X (gfx1250) — compile-verified
//
#include <hip/hip_runtime.h>

#define NPTS 8192
#define DIMS 512
#define LOG2PI 1.8378770664093453f
#define ASTR 516   // LDS row stride in floats: 512 + 4 pad -> conflict-free b64 frag reads
#define JHALF (NPTS / 2)

typedef __attribute__((ext_vector_type(2))) float v2f;
typedef __attribute__((ext_vector_type(8))) float v8f;

// Async global -> LDS copy, 16 bytes per lane. VDST operand = LDS byte offset,
// VADDR = 64-bit global address, SADDR = off. Tracked by ASYNCcnt.
__device__ __forceinline__ void async_b128(unsigned int lds_byte_off, const float* g) {
  asm volatile("global_load_async_to_lds_b128 %0, %1, off"
               :: "v"(lds_byte_off), "v"((unsigned long long)(uintptr_t)g)
               : "memory");
}
__device__ __forceinline__ void wait_async0() {
  asm volatile("s_wait_asynccnt 0x0" ::: "memory");
}

// ---------------- Kernel 1: row squared norms (wave per row) ----------------
__global__ __launch_bounds__(256) void sq_kernel(const float4* __restrict__ x4,
                                                 float* __restrict__ sq) {
  const int lane = threadIdx.x & 31;
  const int wid  = threadIdx.x >> 5;
  const int row  = blockIdx.x * 8 + wid;
  float s = 0.f;
#pragma unroll
  for (int i = 0; i < 4; ++i) {
    float4 v = x4[(size_t)row * 128 + lane + i * 32];
    s += v.x * v.x + v.y * v.y + v.z * v.z + v.w * v.w;
  }
#pragma unroll
  for (int off = 16; off > 0; off >>= 1) s += __shfl_xor(s, off, 32);
  if (lane == 0) sq[row] = s;
}

// ------- Kernel 2: fused WMMA GEMM + online row-wise logsumexp epilogue -----
// Grid: (NPTS/64 row-blocks) x (2 column halves) = 256 WGs.
// Per block: 64 resident rows (A tile in LDS), stream 32-column B tiles with
// double-buffered async global->LDS copies overlapped with WMMA compute.
// Emits per-row raw partials (ib_sum, kde_max, kde_sum) for its column half.
__global__ __launch_bounds__(256) void ib_kernel(
    const float* __restrict__ x, const float* __restrict__ sq,
    const float* __restrict__ nlv, const float* __restrict__ klv,
    float* __restrict__ rib, float* __restrict__ rkm, float* __restrict__ rks) {
  extern __shared__ float smem[];
  // LDS map (floats): A[64*ASTR] | B0[32*ASTR] | B1[32*ASTR] | SQ[8192]
  const unsigned int ldsB0 = 64u * ASTR * 4u;          // byte offsets
  const unsigned int ldsB1 = 96u * ASTR * 4u;
  const unsigned int ldsSQ = 128u * ASTR * 4u;
  float* sA  = smem;
  float* sSQ = smem + 128 * ASTR;

  const int tid   = threadIdx.x;
  const int lane  = tid & 31;
  const int wid   = tid >> 5;    // 0..7
  const int waveM = wid >> 1;    // 0..3  (row group of 16)
  const int waveN = wid & 1;     // 0..1  (col group of 16)
  const int hh    = lane >> 4;   // half-wave: K pair select / row+8
  const int l16   = lane & 15;
  const int rowbase = blockIdx.x * 64;
  const int cs      = blockIdx.y;          // column half: 0 or 1
  const int jb0     = cs * JHALF;

  // ---- prologue: async-stage A tile (64x512), sq[8192], and B buffer 0 ----
#pragma unroll
  for (int i = 0; i < 32; ++i) {           // A: 8192 float4, 32 per thread
    int idx = tid + i * 256;
    int r = idx >> 7, c4 = idx & 127;
    async_b128((unsigned int)(r * ASTR + c4 * 4) * 4u,
               x + (size_t)(rowbase + r) * DIMS + c4 * 4);
  }
#pragma unroll
  for (int i = 0; i < 8; ++i) {            // sq: 2048 float4, 8 per thread
    int idx = tid + i * 256;
    async_b128(ldsSQ + (unsigned int)idx * 16u, sq + idx * 4);
  }
#pragma unroll
  for (int i = 0; i < 16; ++i) {           // B0: 4096 float4, 16 per thread
    int idx = tid + i * 256;
    int r = idx >> 7, c4 = idx & 127;
    async_b128(ldsB0 + (unsigned int)(r * ASTR + c4 * 4) * 4u,
               x + (size_t)(jb0 + r) * DIMS + c4 * 4);
  }
  wait_async0();
  __syncthreads();

  const float nl  = nlv[0], kl = klv[0];
  const float tv  = __expf(nl) + __expf(kl);
  const float nib = -0.5f / tv;          // IB exponent coefficient
  const float nkd = -0.5f / __expf(kl);  // KDE exponent coefficient

  float sqr[8];
#pragma unroll
  for (int v = 0; v < 8; ++v) sqr[v] = sSQ[rowbase + waveM * 16 + hh * 8 + v];

  float ib_s[8], km[8], ks[8];
#pragma unroll
  for (int v = 0; v < 8; ++v) { ib_s[v] = 0.f; km[v] = -3.0e38f; ks[v] = 0.f; }

  // Fragment base pointers (ISA f32 A 16x4 / B 4x16 layouts):
  // lane holds (m|n)=l16, K offset 2*hh, consecutive K pair in 2 VGPRs.
  const float* pA = &sA[(waveM * 16 + l16) * ASTR + 2 * hh];
  const unsigned int fragB = (unsigned int)((waveN * 16 + l16) * ASTR + 2 * hh);

  for (int j = 0; j < JHALF; j += 32) {
    const int jb  = jb0 + j;
    const int buf = (j >> 5) & 1;
    const float* pB = smem + (64 + 32 * buf) * ASTR + fragB;

    // issue async loads for the NEXT tile into the other buffer
    if (j + 32 < JHALF) {
      const unsigned int nbase = (buf ? ldsB0 : ldsB1);
#pragma unroll
      for (int i = 0; i < 16; ++i) {
        int idx = tid + i * 256;
        int r = idx >> 7, c4 = idx & 127;
        async_b128(nbase + (unsigned int)(r * ASTR + c4 * 4) * 4u,
                   x + (size_t)(jb + 32 + r) * DIMS + c4 * 4);
      }
    }

    // ---- compute on current buffer: 128 x v_wmma_f32_16x16x4_f32 ----
    v8f c = {};
#pragma unroll 8
    for (int k = 0; k < DIMS; k += 4) {
      v2f a = *(const v2f*)(pA + k);
      v2f b = *(const v2f*)(pB + k);
      c = __builtin_amdgcn_wmma_f32_16x16x4_f32(
          false, a, false, b, (short)0, c, false, false);
    }

    // ---- fused epilogue: distances + online logsumexp ----
    const int   gn   = jb + waveN * 16 + l16;
    const float scol = sSQ[gn];
#pragma unroll
    for (int v = 0; v < 8; ++v) {
      int   gm   = rowbase + waveM * 16 + hh * 8 + v;
      float dist = sqr[v] + scol - 2.0f * c[v];
      ib_s[v] += __expf(dist * nib);              // args <= ~0, no overflow
      float aa = (gm == gn) ? -3.0e38f : dist * nkd;
      float nm = fmaxf(km[v], aa);
      ks[v] = ks[v] * __expf(km[v] - nm) + __expf(aa - nm);
      km[v] = nm;
    }

    wait_async0();      // next tile landed in LDS
    __syncthreads();    // everyone done computing + loading -> swap buffers
  }

  // ---- cross-lane / cross-wave merge (reuse B buffers as scratch) ----
  float* Pib = smem + 64 * ASTR;       // 64 rows x 32 partial slots
  float* Pkm = Pib + 64 * 32;
  float* Pks = Pkm + 64 * 32;
#pragma unroll
  for (int v = 0; v < 8; ++v) {
    int wr = waveM * 16 + hh * 8 + v;  // 0..63 row in block
    int wc = waveN * 16 + l16;         // 0..31 partial slot
    Pib[wr * 32 + wc] = ib_s[v];
    Pkm[wr * 32 + wc] = km[v];
    Pks[wr * 32 + wc] = ks[v];
  }
  __syncthreads();
  if (tid < 64) {
    float sib = 0.f, M = -3.0e38f, S = 0.f;
    for (int p = 0; p < 32; ++p) {
      sib += Pib[tid * 32 + p];
      float m2 = Pkm[tid * 32 + p], s2 = Pks[tid * 32 + p];
      float nm = fmaxf(M, m2);
      S = S * __expf(M - nm) + s2 * __expf(m2 - nm);
      M = nm;
    }
    int r = rowbase + tid;
    rib[cs * NPTS + r] = sib;   // raw partials; logs taken after cross-half merge
    rkm[cs * NPTS + r] = M;
    rks[cs * NPTS + r] = S;
  }
}

// ------ Kernel 3: merge column halves per row + deterministic block sums ----
__global__ __launch_bounds__(256) void merge_kernel(const float* __restrict__ rib,
                                                    const float* __restrict__ rkm,
                                                    const float* __restrict__ rks,
                                                    float* __restrict__ pib,
                                                    float* __restrict__ pkde) {
  __shared__ float s1[256];
  __shared__ float s2[256];
  const int t = threadIdx.x;
  const int r = blockIdx.x * 256 + t;

  float lp_ib, lp_kde;
  {
    float sib = rib[r] + rib[NPTS + r];
    float m0 = rkm[r], m1 = rkm[NPTS + r];
    float q0 = rks[r], q1 = rks[NPTS + r];
    float M = fmaxf(m0, m1);
    float S = q0 * __expf(m0 - M) + q1 * __expf(m1 - M);
    lp_ib  = __logf(sib);     // per-row logsumexp (IB), row max = 0
    lp_kde = M + __logf(S);   // per-row logsumexp (KDE)
  }
  s1[t] = lp_ib;
  s2[t] = lp_kde;
  __syncthreads();
  for (int off = 128; off > 0; off >>= 1) {
    if (t < off) { s1[t] += s1[t + off]; s2[t] += s2[t + off]; }
    __syncthreads();
  }
  if (t == 0) { pib[blockIdx.x] = s1[0]; pkde[blockIdx.x] = s2[0]; }
}

// ---------------- Kernel 4: x + exp(0.5*nl) * eps ---------------------------
__global__ __launch_bounds__(256) void noise_kernel(const float4* __restrict__ x4,
                                                    const float4* __restrict__ e4,
                                                    const float* __restrict__ nlv,
                                                    float4* __restrict__ o4) {
  size_t i = (size_t)blockIdx.x * 256 + threadIdx.x;
  float s = expf(0.5f * nlv[0]);
  float4 xv = x4[i], ev = e4[i], r;
  r.x = xv.x + s * ev.x; r.y = xv.y + s * ev.y;
  r.z = xv.z + s * ev.z; r.w = xv.w + s * ev.w;
  o4[i] = r;
}

// ---------------- Kernel 5: final scalars (fixed-order reduction) -----------
__global__ __launch_bounds__(32) void fin_kernel(const float* __restrict__ pib,
                                                 const float* __restrict__ pkde,
                                                 const float* __restrict__ nlv,
                                                 const float* __restrict__ klv,
                                                 float* __restrict__ out2) {
  if (threadIdx.x == 0) {
    float a = 0.f, b = 0.f;
    for (int i = 0; i < NPTS / 256; ++i) { a += pib[i]; b += pkde[i]; }
    float nl = nlv[0], kl = klv[0];
    float tv = expf(nl) + expf(kl);
    float normc = 0.5f * DIMS * (LOG2PI + logf(tv));
    float mean_ib = a / (float)NPTS;
    float h = -(mean_ib - logf((float)NPTS) - normc);
    float hcond = 0.5f * DIMS * (LOG2PI + nl);
    float normk = 0.5f * DIMS * (LOG2PI + kl);
    float mean_kde = b / (float)NPTS;
    out2[0] = h - hcond;                                      // ib_cost
    out2[1] = -(mean_kde - logf((float)NPTS - 1.0f) - normk); // kde_loo_loss
  }
}

// ---------------------------------------------------------------------------
extern "C" void kernel_launch(void* const* d_in, const int* in_sizes, int n_in,
                              void* d_out, int out_size, void* d_ws, size_t ws_size,
                              hipStream_t stream) {
  const float* x   = (const float*)d_in[0];
  const float* nlv = (const float*)d_in[1];
  const float* klv = (const float*)d_in[2];
  const float* eps = (const float*)d_in[3];
  float* out = (float*)d_out;

  // workspace layout (floats)
  float* sqv  = (float*)d_ws;         // NPTS
  float* rib  = sqv + NPTS;           // 2*NPTS
  float* rkm  = rib + 2 * NPTS;       // 2*NPTS
  float* rks  = rkm + 2 * NPTS;       // 2*NPTS
  float* pib  = rks + 2 * NPTS;       // NPTS/256
  float* pkde = pib + NPTS / 256;     // NPTS/256

  sq_kernel<<<NPTS / 8, 256, 0, stream>>>((const float4*)x, sqv);

  // LDS: A(64*ASTR) + B0/B1(2*32*ASTR) + sq(8192) floats = 296,960 B < 320KB/WGP
  size_t lds_bytes = (size_t)(128 * ASTR + NPTS) * sizeof(float);
  dim3 grid(NPTS / 64, 2);   // 128 row-blocks x 2 column halves = 256 WGs
  ib_kernel<<<grid, 256, lds_bytes, stream>>>(x, sqv, nlv, klv, rib, rkm, rks);

  merge_kernel<<<NPTS / 256, 256, 0, stream>>>(rib, rkm, rks, pib, pkde);

  noise_kernel<<<(NPTS * DIMS / 4) / 256, 256, 0, stream>>>(
      (const float4*)x, (const float4*)eps, nlv, (float4*)out);

  fin_kernel<<<1, 32, 0, stream>>>(pib, pkde, nlv, klv, out + (size_t)NPTS * DIMS);
}